// Qlora_block_split_70360154243683
// MI455X (gfx1250) — compile-verified
//
#include <hip/hip_runtime.h>
#include <stdint.h>
#include <math.h>

// NF4 block-quantize + dequantize round-trip for (rows, 4096) f32.
// One workgroup per row; quant block = 64 elems = 4 adjacent wave32 lanes.
// Nearest-level search via an exact 64-cell LUT: each cell of [-1,1] holds at
// most one argmin breakpoint (min breakpoint gap 0.0847 > 1/32), so
// level(xn) = (xn > mid[cell]) ? hi[cell] : lo[cell]  -- bit-exact argmin.

#define COLS   4096
#define QBLK   64
#define TPB    256
#define EPT    16            // COLS / TPB floats per thread (= 1/4 quant block)
#define EPS_F  1e-8f
#define NCELL  64

typedef float v4f __attribute__((ext_vector_type(4)));

// fp16 round-trip (constant-folded): reference table is ndtri-quantiles cast to f16
__device__ __forceinline__ float f16r(float v) { return (float)(_Float16)v; }

// CDNA5 async global->LDS copy, tracked by ASYNCcnt (CDNA5 ISA)
__device__ __forceinline__ void async_ld_b128(uint32_t lds_off, const void* gaddr) {
  asm volatile("global_load_async_to_lds_b128 %0, %1, off"
               :: "v"(lds_off), "v"(gaddr)
               : "memory");
}

__global__ __launch_bounds__(TPB)
void nf4_roundtrip_kernel(const float* __restrict__ in, float* __restrict__ out) {
  __shared__ __align__(16) float s_row[COLS];        // 16 KB row staging
  __shared__ float s_scale[COLS / QBLK];             // 64 block scales
  __shared__ float s_mid[NCELL];                     // breakpoint per cell (+inf if none)
  __shared__ float s_lo[NCELL];                      // level left of breakpoint
  __shared__ float s_hi[NCELL];                      // level right of breakpoint

  const int tid   = threadIdx.x;
  const size_t rb = (size_t)blockIdx.x * COLS;
  const float* gsrc = in + rb + tid * EPT;

  // ---- async global -> LDS: 4 x b128 per thread = full 16 KB row ----
  uint32_t lds0 = (uint32_t)(uintptr_t)&s_row[tid * EPT];
  #pragma unroll
  for (int i = 0; i < 4; ++i)
    async_ld_b128(lds0 + 16u * (uint32_t)i, (const void*)(gsrc + 4 * i));

  // ---- build the 64-cell argmin LUT while the DMA is in flight ----
  // NF4 table: normalized ndtri((i+.5)/16) rounded to fp16 (folded to literals).
  if (tid < NCELL) {
    const float T[16] = {
      f16r(-1.0f),       f16r(-0.7075688f), f16r(-0.5422091f), f16r(-0.4168190f),
      f16r(-0.3109041f), f16r(-0.2159463f), f16r(-0.1273412f), f16r(-0.0420957f),
      f16r( 0.0420957f), f16r( 0.1273412f), f16r( 0.2159463f), f16r( 0.3109041f),
      f16r( 0.4168190f), f16r( 0.5422091f), f16r( 0.7075688f), f16r( 1.0f) };
    const float cs = -1.0f + (float)tid * (1.0f / 32.0f);   // cell start
    const float ce = cs + (1.0f / 32.0f);                   // cell end
    float lo = T[0], hi = T[0], mid = INFINITY;
    #pragma unroll
    for (int j = 0; j < 15; ++j) {
      const float M = 0.5f * (T[j] + T[j + 1]);  // exact f32 (fp16 sums are exact)
      const float D = T[j + 1] - T[j];
      if (cs > M) lo += D;            // level at cell start ('>' = argmin tie rule)
      if (ce > M) hi += D;            // level just below cell end
      if (cs <= M && M < ce) mid = M; // breakpoint owned by this cell
    }
    s_mid[tid] = mid; s_lo[tid] = lo; s_hi[tid] = hi;
  }

  asm volatile("s_wait_asynccnt 0" ::: "memory");
  __syncthreads();

  // ---- pull my 16 values into registers (ds_load_b128 x4) ----
  float x[EPT];
  const float4* lp = reinterpret_cast<const float4*>(&s_row[tid * EPT]);
  #pragma unroll
  for (int i = 0; i < 4; ++i) {
    float4 v = lp[i];
    x[4*i+0] = v.x; x[4*i+1] = v.y; x[4*i+2] = v.z; x[4*i+3] = v.w;
  }

  // ---- quant-block min/max: thread-local then across 4 adjacent lanes ----
  float vmin = x[0], vmax = x[0];
  #pragma unroll
  for (int i = 1; i < EPT; ++i) { vmin = fminf(vmin, x[i]); vmax = fmaxf(vmax, x[i]); }
  #pragma unroll
  for (int m = 1; m <= 2; m <<= 1) {
    vmin = fminf(vmin, __shfl_xor(vmin, m, 32));
    vmax = fmaxf(vmax, __shfl_xor(vmax, m, 32));
  }
  const float scale = vmax - vmin;
  if ((tid & 3) == 0) s_scale[tid >> 2] = scale;
  __syncthreads();

  // ---- per-row min/max over the 64 block scales (wave-local xor reduction) ----
  const int l = tid & 31;
  float s0 = s_scale[l], s1 = s_scale[l + 32];
  float smin = fminf(s0, s1), smax = fmaxf(s0, s1);
  #pragma unroll
  for (int off = 16; off >= 1; off >>= 1) {
    smin = fminf(smin, __shfl_xor(smin, off, 32));
    smax = fmaxf(smax, __shfl_xor(smax, off, 32));
  }

  // ---- 8-bit scale quantization; jnp.round == round-to-nearest-even ----
  const float sq  = rintf((scale - smin) / (smax - smin + EPS_F) * 255.0f);
  const float sdq = smin + sq * (1.0f / 255.0f) * (smax - smin);

  const float inv = 2.0f / (scale + EPS_F);   // xn = (x - vmin)*inv - 1
  const float a   = 0.5f * sdq;               // out = w*a + (a + vmin)
  const float b   = a + vmin;

  v4f o[4];
  float* of = reinterpret_cast<float*>(o);
  #pragma unroll
  for (int i = 0; i < EPT; ++i) {
    const float xn = fmaf(x[i] - vmin, inv, -1.0f);
    const float fi = fmaf(xn, 32.0f, 32.0f);       // cell coordinate in [0,64)
    int idx = (int)fi;                             // trunc == floor for fi > -1
    idx = idx < 0 ? 0 : (idx > NCELL - 1 ? NCELL - 1 : idx);   // v_med3_i32
    const float mid = s_mid[idx];
    const float lo  = s_lo[idx];
    const float hi  = s_hi[idx];
    const float w   = (xn > mid) ? hi : lo;        // exact argmin w/ tie rule
    of[i] = fmaf(w, a, b);
  }

  // ---- streaming (non-temporal) b128 stores: output is never re-read ----
  v4f* gout = (v4f*)(out + rb + tid * EPT);
  #pragma unroll
  for (int i = 0; i < 4; ++i) __builtin_nontemporal_store(o[i], gout + i);
}

extern "C" void kernel_launch(void* const* d_in, const int* in_sizes, int n_in,
                              void* d_out, int out_size, void* d_ws, size_t ws_size,
                              hipStream_t stream) {
  (void)n_in; (void)out_size; (void)d_ws; (void)ws_size;
  const float* in = (const float*)d_in[0];
  float* out      = (float*)d_out;
  const int rows  = in_sizes[0] / COLS;   // (4*1024*4096)/4096 = 4096 rows
  nf4_roundtrip_kernel<<<dim3(rows), dim3(TPB), 0, stream>>>(in, out);
}